// GConvGRU_21062519620361
// MI455X (gfx1250) — compile-verified
//
#include <hip/hip_runtime.h>
#include <hip/hip_bf16.h>

#define NN 50000
#define NE 800000
#define C  64
#define TILES (NN / 16)          // 3125, exact

typedef __attribute__((ext_vector_type(2))) float v2f;
typedef __attribute__((ext_vector_type(8))) float v8f;

// ---------------- elementwise / sparse kernels ----------------

__global__ void zero_f32(float* p, int n) {
  int i = blockIdx.x * blockDim.x + threadIdx.x;
  if (i < n) p[i] = 0.0f;
}

__global__ void deg_kernel(const int* __restrict__ ei, const float* __restrict__ w,
                           float* __restrict__ deg) {
  int e = blockIdx.x * blockDim.x + threadIdx.x;
  if (e < NE) atomicAdd(&deg[ei[e]], w[e]);
}

__global__ void lw_kernel(const int* __restrict__ ei, const float* __restrict__ w,
                          const float* __restrict__ deg, float* __restrict__ lw) {
  int e = blockIdx.x * blockDim.x + threadIdx.x;
  if (e >= NE) return;
  int s = ei[e], d = ei[NE + e];
  float ds = deg[s], dd = deg[d];
  float is = ds > 0.0f ? rsqrtf(fmaxf(ds, 1e-12f)) : 0.0f;
  float id = dd > 0.0f ? rsqrtf(fmaxf(dd, 1e-12f)) : 0.0f;
  lw[e] = -is * w[e] * id;
}

// y = (t0 ? -t0 : 0), vectorized float4; n4 = NN*C/4
__global__ void init_dst(float4* __restrict__ y, const float4* __restrict__ t0, int n4) {
  int i = blockIdx.x * blockDim.x + threadIdx.x;
  if (i >= n4) return;
  if (t0) {
    float4 v = t0[i];
    y[i] = make_float4(-v.x, -v.y, -v.z, -v.w);
  } else {
    y[i] = make_float4(0.f, 0.f, 0.f, 0.f);
  }
}

// Y[dst] += scale*lw[e]*T[src], 4 channels per thread (E*16 threads)
__global__ void scatter_spmm(const int* __restrict__ ei, const float* __restrict__ lw,
                             const float* __restrict__ t, float* __restrict__ y,
                             float scale) {
  int idx = blockIdx.x * blockDim.x + threadIdx.x;
  if (idx >= NE * 16) return;
  int e = idx >> 4;
  int c = (idx & 15) << 2;
  int s = ei[e];
  int d = ei[NE + e];
  float wv = scale * lw[e];
  const float4 tv = *(const float4*)(t + (size_t)s * C + c);
  float* yp = y + (size_t)d * C + c;
  atomicAdd(yp + 0, wv * tv.x);
  atomicAdd(yp + 1, wv * tv.y);
  atomicAdd(yp + 2, wv * tv.z);
  atomicAdd(yp + 3, wv * tv.w);
}

// ---------------- weight repack into WMMA B-fragment order ----------------
// For slice s (64x64 row-major, rows=K dim, cols=N dim), fragment f = k0idx*4+nt:
//   Wp[s][f*64 + lane*2 + c] = W[s][(k0idx*4 + (lane>>4)*2 + c)*64 + nt*16 + (lane&15)]
// so a lane's v2f B fragment is one aligned contiguous b64, coalesced across lanes.
__global__ void repack_w(const float* __restrict__ W, float* __restrict__ Wp, int nslices) {
  int i = blockIdx.x * blockDim.x + threadIdx.x;
  if (i >= nslices * 4096) return;
  int s = i >> 12;
  int r = i & 4095;
  int f = r >> 6;
  int t = r & 63;
  int lane = t >> 1, c = t & 1;
  int k0idx = f >> 2, nt = f & 3;
  int row = k0idx * 4 + (lane >> 4) * 2 + c;
  int col = nt * 16 + (lane & 15);
  Wp[(size_t)s * 4096 + r] = W[(size_t)s * 4096 + (size_t)row * 64 + col];
}

// ---------------- fused WMMA gate kernels ----------------
// Fragment helpers: packed slice base -> per-lane fragment pointer
__device__ __forceinline__ const float* frag_base(const float* Wp_slice, int k0idx, int lane) {
  return Wp_slice + (size_t)k0idx * 256 + lane * 2;
}

// Computes BOTH: Z = sigmoid(sum Tx_k@Wxz[k] + Th_k@Whz[k] + bz)
//          and  HR = H * sigmoid(sum Tx_k@Wxr[k] + Th_k@Whr[k] + br)
__global__ __launch_bounds__(128)
void gate2_wmma(const float* __restrict__ tx0, const float* __restrict__ tx1,
                const float* __restrict__ tx2, const float* __restrict__ th0,
                const float* __restrict__ th1, const float* __restrict__ th2,
                const float* __restrict__ Wpxz, const float* __restrict__ Wphz,
                const float* __restrict__ Wpxr, const float* __restrict__ Wphr,
                const float* __restrict__ bxz, const float* __restrict__ bhz,
                const float* __restrict__ bxr, const float* __restrict__ bhr,
                const float* __restrict__ H,
                float* __restrict__ Zout, float* __restrict__ HRout) {
  const int wave = threadIdx.x >> 5;
  const int lane = threadIdx.x & 31;
  const int tile = blockIdx.x * 4 + wave;       // wave-uniform
  if (tile >= TILES) return;                    // whole wave exits together

  const int m0 = tile * 16;
  const int ln = lane & 15;
  const int hi = lane >> 4;

  v8f az0 = {}, az1 = {}, az2 = {}, az3 = {};
  v8f ar0 = {}, ar1 = {}, ar2 = {}, ar3 = {};

  const float* Ts[6] = {tx0, tx1, tx2, th0, th1, th2};
#pragma unroll
  for (int term = 0; term < 6; ++term) {
    const float* T = Ts[term];
    const int slice = (term < 3) ? term : (term - 3);
    const float* Wz = ((term < 3) ? Wpxz : Wphz) + (size_t)slice * 4096;
    const float* Wr = ((term < 3) ? Wpxr : Wphr) + (size_t)slice * 4096;
    const float* arow = T + (size_t)(m0 + ln) * C;
    for (int k0idx = 0; k0idx < 16; ++k0idx) {
      const int kb = k0idx * 4 + hi * 2;
      v2f a;
      a.x = arow[kb];
      a.y = arow[kb + 1];
      const float* fz = frag_base(Wz, k0idx, lane);
      const float* fr = frag_base(Wr, k0idx, lane);
      v2f bz0 = *(const v2f*)(fz);
      v2f bz1 = *(const v2f*)(fz + 64);
      v2f bz2 = *(const v2f*)(fz + 128);
      v2f bz3 = *(const v2f*)(fz + 192);
      v2f br0 = *(const v2f*)(fr);
      v2f br1 = *(const v2f*)(fr + 64);
      v2f br2 = *(const v2f*)(fr + 128);
      v2f br3 = *(const v2f*)(fr + 192);
      az0 = __builtin_amdgcn_wmma_f32_16x16x4_f32(false, a, false, bz0, (short)0, az0, false, false);
      az1 = __builtin_amdgcn_wmma_f32_16x16x4_f32(false, a, false, bz1, (short)0, az1, false, false);
      az2 = __builtin_amdgcn_wmma_f32_16x16x4_f32(false, a, false, bz2, (short)0, az2, false, false);
      az3 = __builtin_amdgcn_wmma_f32_16x16x4_f32(false, a, false, bz3, (short)0, az3, false, false);
      ar0 = __builtin_amdgcn_wmma_f32_16x16x4_f32(false, a, false, br0, (short)0, ar0, false, false);
      ar1 = __builtin_amdgcn_wmma_f32_16x16x4_f32(false, a, false, br1, (short)0, ar1, false, false);
      ar2 = __builtin_amdgcn_wmma_f32_16x16x4_f32(false, a, false, br2, (short)0, ar2, false, false);
      ar3 = __builtin_amdgcn_wmma_f32_16x16x4_f32(false, a, false, br3, (short)0, ar3, false, false);
    }
  }

#pragma unroll
  for (int nt = 0; nt < 4; ++nt) {
    v8f vz = (nt == 0) ? az0 : (nt == 1) ? az1 : (nt == 2) ? az2 : az3;
    v8f vr = (nt == 0) ? ar0 : (nt == 1) ? ar1 : (nt == 2) ? ar2 : ar3;
    const int col = nt * 16 + ln;
    const float biasz = bxz[col] + bhz[col];
    const float biasr = bxr[col] + bhr[col];
#pragma unroll
    for (int r = 0; r < 8; ++r) {
      const int row = m0 + r + hi * 8;
      const size_t off = (size_t)row * C + col;
      const float z = 1.0f / (1.0f + __expf(-(vz[r] + biasz)));
      const float rr = 1.0f / (1.0f + __expf(-(vr[r] + biasr)));
      Zout[off] = z;
      HRout[off] = H[off] * rr;
    }
  }
}

// H_new = Z*H + (1-Z)*tanh(sum Tx_k@Wxh[k] + THR_k@Whh[k] + b)
__global__ __launch_bounds__(128)
void gate_final_wmma(const float* __restrict__ tx0, const float* __restrict__ tx1,
                     const float* __restrict__ tx2, const float* __restrict__ th0,
                     const float* __restrict__ th1, const float* __restrict__ th2,
                     const float* __restrict__ Wpxh, const float* __restrict__ Wphh,
                     const float* __restrict__ bxh, const float* __restrict__ bhh,
                     const float* __restrict__ Zg, const float* __restrict__ Hg,
                     float* __restrict__ out) {
  const int wave = threadIdx.x >> 5;
  const int lane = threadIdx.x & 31;
  const int tile = blockIdx.x * 4 + wave;
  if (tile >= TILES) return;

  const int m0 = tile * 16;
  const int ln = lane & 15;
  const int hi = lane >> 4;

  v8f acc0 = {}, acc1 = {}, acc2 = {}, acc3 = {};

  const float* Ts[6] = {tx0, tx1, tx2, th0, th1, th2};
#pragma unroll
  for (int term = 0; term < 6; ++term) {
    const float* T = Ts[term];
    const int slice = (term < 3) ? term : (term - 3);
    const float* W = ((term < 3) ? Wpxh : Wphh) + (size_t)slice * 4096;
    const float* arow = T + (size_t)(m0 + ln) * C;
    for (int k0idx = 0; k0idx < 16; ++k0idx) {
      const int kb = k0idx * 4 + hi * 2;
      v2f a;
      a.x = arow[kb];
      a.y = arow[kb + 1];
      const float* fb = frag_base(W, k0idx, lane);
      v2f b0 = *(const v2f*)(fb);
      v2f b1 = *(const v2f*)(fb + 64);
      v2f b2 = *(const v2f*)(fb + 128);
      v2f b3 = *(const v2f*)(fb + 192);
      acc0 = __builtin_amdgcn_wmma_f32_16x16x4_f32(false, a, false, b0, (short)0, acc0, false, false);
      acc1 = __builtin_amdgcn_wmma_f32_16x16x4_f32(false, a, false, b1, (short)0, acc1, false, false);
      acc2 = __builtin_amdgcn_wmma_f32_16x16x4_f32(false, a, false, b2, (short)0, acc2, false, false);
      acc3 = __builtin_amdgcn_wmma_f32_16x16x4_f32(false, a, false, b3, (short)0, acc3, false, false);
    }
  }

#pragma unroll
  for (int nt = 0; nt < 4; ++nt) {
    v8f v = (nt == 0) ? acc0 : (nt == 1) ? acc1 : (nt == 2) ? acc2 : acc3;
    const int col = nt * 16 + ln;
    const float bias = bxh[col] + bhh[col];
#pragma unroll
    for (int r = 0; r < 8; ++r) {
      const int row = m0 + r + hi * 8;
      const size_t off = (size_t)row * C + col;
      const float ht = tanhf(v[r] + bias);
      const float z = Zg[off];
      const float h = Hg[off];
      out[off] = z * h + (1.0f - z) * ht;
    }
  }
}

// ---------------- launcher ----------------

extern "C" void kernel_launch(void* const* d_in, const int* in_sizes, int n_in,
                              void* d_out, int out_size, void* d_ws, size_t ws_size,
                              hipStream_t stream) {
  const float* X    = (const float*)d_in[0];
  const int*   ei   = (const int*)d_in[1];      // [2,E] row-major: src then dst
  const float* ew   = (const float*)d_in[2];
  const float* H    = (const float*)d_in[3];
  const float* W_xz = (const float*)d_in[4];
  const float* b_xz = (const float*)d_in[5];
  const float* W_hz = (const float*)d_in[6];
  const float* b_hz = (const float*)d_in[7];
  const float* W_xr = (const float*)d_in[8];
  const float* b_xr = (const float*)d_in[9];
  const float* W_hr = (const float*)d_in[10];
  const float* b_hr = (const float*)d_in[11];
  const float* W_xh = (const float*)d_in[12];
  const float* b_xh = (const float*)d_in[13];
  const float* W_hh = (const float*)d_in[14];
  const float* b_hh = (const float*)d_in[15];
  float* out = (float*)d_out;

  float* ws  = (float*)d_ws;
  float* deg = ws;  ws += NN;
  float* lw  = ws;  ws += NE;
  const size_t NC = (size_t)NN * C;
  float* TX1 = ws;  ws += NC;
  float* TX2 = ws;  ws += NC;
  float* TH1 = ws;  ws += NC;   // reused as THR1 later
  float* TH2 = ws;  ws += NC;   // reused as THR2 later
  float* Zb  = ws;  ws += NC;
  float* HRb = ws;  ws += NC;
  // packed weights: 6 sets x 3 slices x 4096 floats
  float* Wpxz = ws; ws += 3 * 4096;
  float* Wphz = ws; ws += 3 * 4096;
  float* Wpxr = ws; ws += 3 * 4096;
  float* Wphr = ws; ws += 3 * 4096;
  float* Wpxh = ws; ws += 3 * 4096;
  float* Wphh = ws; ws += 3 * 4096;

  const int n4 = (int)(NC / 4);
  dim3 b256(256);
  dim3 gN((NN + 255) / 256);
  dim3 gE((NE + 255) / 256);
  dim3 gNC4((n4 + 255) / 256);
  dim3 gScat((NE * 16 + 255) / 256);
  dim3 gGate((TILES + 3) / 4);
  dim3 bGate(128);
  dim3 gRep((3 * 4096 + 255) / 256);

  // Weight repack (fragment-major for b64 B loads)
  repack_w<<<gRep, b256, 0, stream>>>(W_xz, Wpxz, 3);
  repack_w<<<gRep, b256, 0, stream>>>(W_hz, Wphz, 3);
  repack_w<<<gRep, b256, 0, stream>>>(W_xr, Wpxr, 3);
  repack_w<<<gRep, b256, 0, stream>>>(W_hr, Wphr, 3);
  repack_w<<<gRep, b256, 0, stream>>>(W_xh, Wpxh, 3);
  repack_w<<<gRep, b256, 0, stream>>>(W_hh, Wphh, 3);

  // Laplacian weights
  zero_f32<<<gN, b256, 0, stream>>>(deg, NN);
  deg_kernel<<<gE, b256, 0, stream>>>(ei, ew, deg);
  lw_kernel<<<gE, b256, 0, stream>>>(ei, ew, deg, lw);

  // Chebyshev propagation of X: TX1 = L X ; TX2 = 2 L TX1 - X
  init_dst<<<gNC4, b256, 0, stream>>>((float4*)TX1, (const float4*)nullptr, n4);
  scatter_spmm<<<gScat, b256, 0, stream>>>(ei, lw, X, TX1, 1.0f);
  init_dst<<<gNC4, b256, 0, stream>>>((float4*)TX2, (const float4*)X, n4);
  scatter_spmm<<<gScat, b256, 0, stream>>>(ei, lw, TX1, TX2, 2.0f);

  // Chebyshev propagation of H
  init_dst<<<gNC4, b256, 0, stream>>>((float4*)TH1, (const float4*)nullptr, n4);
  scatter_spmm<<<gScat, b256, 0, stream>>>(ei, lw, H, TH1, 1.0f);
  init_dst<<<gNC4, b256, 0, stream>>>((float4*)TH2, (const float4*)H, n4);
  scatter_spmm<<<gScat, b256, 0, stream>>>(ei, lw, TH1, TH2, 2.0f);

  // Fused Z and H*R gates (shares all six T-matrix reads)
  gate2_wmma<<<gGate, bGate, 0, stream>>>(X, TX1, TX2, H, TH1, TH2,
                                          Wpxz, Wphz, Wpxr, Wphr,
                                          b_xz, b_hz, b_xr, b_hr,
                                          H, Zb, HRb);

  // Chebyshev propagation of HR (reuse TH1/TH2)
  init_dst<<<gNC4, b256, 0, stream>>>((float4*)TH1, (const float4*)nullptr, n4);
  scatter_spmm<<<gScat, b256, 0, stream>>>(ei, lw, HRb, TH1, 1.0f);
  init_dst<<<gNC4, b256, 0, stream>>>((float4*)TH2, (const float4*)HRb, n4);
  scatter_spmm<<<gScat, b256, 0, stream>>>(ei, lw, TH1, TH2, 2.0f);

  // H_new = Z*H + (1-Z)*tanh(cheb(X;Wxh) + cheb(HR;Whh))
  gate_final_wmma<<<gGate, bGate, 0, stream>>>(X, TX1, TX2, HRb, TH1, TH2,
                                               Wpxh, Wphh, b_xh, b_hh,
                                               Zb, H, out);
}